// Temporal_Attention_layer_43739946942868
// MI455X (gfx1250) — compile-verified
//
#include <hip/hip_runtime.h>
#include <hip/hip_bf16.h>
#include <math.h>

// Problem sizes
constexpr int NV = 4096;
constexpr int NC = 32;
constexpr int NT = 512;

typedef float v2f __attribute__((ext_vector_type(2)));
typedef float v8f __attribute__((ext_vector_type(8)));

// ---------------------------------------------------------------------------
// f32 WMMA 16x16x4 helpers (wave32).
// ISA fragment layout (CDNA5 ISA):
//   A[m,k]: lane = m + 16*(k>>1), vgpr q = k&1  ->  k = 2*laneHi + q, m = laneLo
//   B[k,n]: symmetric (n = laneLo, k = 2*laneHi + q)
//   D[m,n]: vgpr p holds m = p + 8*laneHi, n = laneLo
// A K-major operand (element [k][m] at p[k*ld + m]) therefore loads as two
// coalesced 64B segments per fragment register.
// ---------------------------------------------------------------------------
__device__ __forceinline__ v2f frag_kmajor(const float* __restrict__ p, int ld,
                                           int k0, int m0) {
  const int lane = threadIdx.x & 31;
  const int lo = lane & 15;
  const int hi = lane >> 4;
  const float* base = p + (size_t)(k0 + 2 * hi) * ld + (m0 + lo);
  v2f f;
  f.x = base[0];        // k = k0 + 2*hi
  f.y = base[ld];       // k = k0 + 2*hi + 1
  return f;
}

__device__ __forceinline__ v8f wmma_f32(v2f a, v2f b, v8f c) {
  return __builtin_amdgcn_wmma_f32_16x16x4_f32(
      /*neg_a=*/false, a, /*neg_b=*/false, b,
      /*c_mod=*/(short)0, c, /*reuse_a=*/false, /*reuse_b=*/false);
}

// ---------------------------------------------------------------------------
// K1: single pass over x (256 MB, HBM-bound kernel of the whole pipeline).
//   Rvt[v*NT + t]      = sum_c x[v,c,t] * U3[c]            (K-major rhs)
//   tmpP[blk][c*NT+t]  = partial over 64 v's of x[v,c,t]*U1[v]
// Each thread owns 2 fixed t-columns -> register accumulators, no races.
// ---------------------------------------------------------------------------
__global__ __launch_bounds__(256) void k1_reduce_x(
    const float* __restrict__ x, const float* __restrict__ U1,
    const float* __restrict__ U3, float* __restrict__ tmpP,
    float* __restrict__ Rvt) {
  __shared__ float u3s[NC];
  const int tid = threadIdx.x;
  if (tid < NC) u3s[tid] = U3[tid];
  __syncthreads();

  float accr[2][NC];
#pragma unroll
  for (int j = 0; j < 2; ++j)
#pragma unroll
    for (int c = 0; c < NC; ++c) accr[j][c] = 0.f;

  const int v0 = blockIdx.x * 64;
  for (int dv = 0; dv < 64; ++dv) {
    const int v = v0 + dv;
    const float u1v = U1[v];
    const float* xv_base = x + (size_t)v * NC * NT;
#pragma unroll
    for (int j = 0; j < 2; ++j) {
      const int t = tid + j * 256;
      float rs = 0.f;
#pragma unroll
      for (int c = 0; c < NC; ++c) {
        const float xv = xv_base[(size_t)c * NT + t];   // coalesced over lanes
        rs = fmaf(xv, u3s[c], rs);
        accr[j][c] = fmaf(xv, u1v, accr[j][c]);
      }
      Rvt[(size_t)v * NT + t] = rs;                     // coalesced
    }
  }

  float* outp = tmpP + (size_t)blockIdx.x * (NC * NT);
#pragma unroll
  for (int j = 0; j < 2; ++j)
#pragma unroll
    for (int c = 0; c < NC; ++c)
      outp[c * NT + tid + j * 256] = accr[j][c];
}

// K1b: deterministic reduction of the 64 per-block partials -> tmpCT[c*NT+t]
__global__ __launch_bounds__(256) void k1b_sum_partials(
    const float* __restrict__ tmpP, float* __restrict__ tmpCT) {
  const int i = blockIdx.x * 256 + threadIdx.x;   // 0 .. NC*NT-1
  float s = 0.f;
  for (int b = 0; b < 64; ++b) s += tmpP[(size_t)b * (NC * NT) + i];
  tmpCT[i] = s;
}

// ---------------------------------------------------------------------------
// K2: lhs (stored K-major): Lvt[v*NT + t] = sum_c U2[c,v] * tmpCT[c,t]
// ---------------------------------------------------------------------------
__global__ __launch_bounds__(256) void k2_lhs(const float* __restrict__ U2,
                                              const float* __restrict__ tmpCT,
                                              float* __restrict__ Lvt) {
  const int v = blockIdx.x;
  float u2r[NC];
#pragma unroll
  for (int c = 0; c < NC; ++c) u2r[c] = U2[(size_t)c * NV + v];  // uniform
#pragma unroll
  for (int j = 0; j < 2; ++j) {
    const int t = threadIdx.x + j * 256;
    float a = 0.f;
#pragma unroll
    for (int c = 0; c < NC; ++c) a = fmaf(u2r[c], tmpCT[c * NT + t], a);
    Lvt[(size_t)v * NT + t] = a;
  }
}

// ---------------------------------------------------------------------------
// K3: VeT[s*NT + t] = V_e[t*NT + s]   (K-major A operand for the E matmul)
// ---------------------------------------------------------------------------
__global__ void k3_transpose(const float* __restrict__ Ve,
                             float* __restrict__ VeT) {
  __shared__ float tile[32][33];
  const int bx = blockIdx.x * 32, by = blockIdx.y * 32;
  const int tx = threadIdx.x, ty = threadIdx.y;  // 32 x 8
#pragma unroll
  for (int j = 0; j < 32; j += 8)
    tile[ty + j][tx] = Ve[(size_t)(by + ty + j) * NT + bx + tx];
  __syncthreads();
#pragma unroll
  for (int j = 0; j < 32; j += 8)
    VeT[(size_t)(bx + ty + j) * NT + by + tx] = tile[tx][ty + j];
}

// ---------------------------------------------------------------------------
// K4: product[t,s] = sum_v rhs[t,v]*lhs[s,v]  (512x512, K=4096)
// 256 waves total; each wave computes a 16(M) x 64(N) tile: A fragment is
// loaded once per K-step and reused across 4 WMMAs.
// ---------------------------------------------------------------------------
__global__ __launch_bounds__(256) void k4_product(const float* __restrict__ Rvt,
                                                  const float* __restrict__ Lvt,
                                                  float* __restrict__ prod) {
  const int wid = threadIdx.x >> 5;
  const int lane = threadIdx.x & 31;
  const int lo = lane & 15, hi = lane >> 4;
  const int g = blockIdx.x * 8 + wid;   // 0..255
  const int t0 = (g >> 3) * 16;         // 32 M tiles
  const int s0 = (g & 7) * 64;          // 8 N groups of 64

  v8f acc0 = {}, acc1 = {}, acc2 = {}, acc3 = {};
  for (int k = 0; k < NV; k += 4) {
    const v2f a = frag_kmajor(Rvt, NT, k, t0);
    const v2f b0 = frag_kmajor(Lvt, NT, k, s0);
    const v2f b1 = frag_kmajor(Lvt, NT, k, s0 + 16);
    const v2f b2 = frag_kmajor(Lvt, NT, k, s0 + 32);
    const v2f b3 = frag_kmajor(Lvt, NT, k, s0 + 48);
    acc0 = wmma_f32(a, b0, acc0);
    acc1 = wmma_f32(a, b1, acc1);
    acc2 = wmma_f32(a, b2, acc2);
    acc3 = wmma_f32(a, b3, acc3);
  }
#pragma unroll
  for (int p = 0; p < 8; ++p) {
    const int t = t0 + p + 8 * hi;
    float* row = prod + (size_t)t * NT + s0 + lo;
    row[0] = acc0[p];
    row[16] = acc1[p];
    row[32] = acc2[p];
    row[48] = acc3[p];
  }
}

// ---------------------------------------------------------------------------
// K5: fused  E[c] = VeT^T @ sigmoid(product + b_e[c])  +  softmax over t.
// Block = (c, 16-wide r stripe); owns the FULL t-column (512) of E:
//   8 waves x 4 M-tiles x 8 acc VGPRs. Sigmoid B-tiles staged in LDS once
//   per 64-deep K chunk and shared by all 8 waves. Softmax entirely on-chip.
// ---------------------------------------------------------------------------
__global__ __launch_bounds__(256) void k5_attn(const float* __restrict__ VeT,
                                               const float* __restrict__ prod,
                                               const float* __restrict__ b_e,
                                               float* __restrict__ out) {
  constexpr int KC = 64;
  __shared__ float sigB[KC * 16];
  __shared__ float red[8][16];

  const int tid = threadIdx.x;
  const int wid = tid >> 5;
  const int lane = tid & 31;
  const int lo = lane & 15, hi = lane >> 4;
  const int c = blockIdx.y;
  const int r0 = blockIdx.x * 16;
  const float* bec = b_e + (size_t)c * NT * NT;
  const int tbase = wid * 64;

  v8f acc[4] = {v8f{}, v8f{}, v8f{}, v8f{}};

  for (int s0 = 0; s0 < NT; s0 += KC) {
    // Stage sigmoid(product + b_e) tile (64 x 16) cooperatively.
    for (int idx = tid; idx < KC * 16; idx += 256) {
      const int kk = idx >> 4, n = idx & 15;
      const size_t off = (size_t)(s0 + kk) * NT + r0 + n;
      const float z = prod[off] + bec[off];
      sigB[idx] = 1.f / (1.f + __expf(-z));
    }
    __syncthreads();

#pragma unroll
    for (int kq = 0; kq < KC / 4; ++kq) {
      v2f b;
      b.x = sigB[(kq * 4 + 2 * hi) * 16 + lo];
      b.y = sigB[(kq * 4 + 2 * hi + 1) * 16 + lo];
#pragma unroll
      for (int i = 0; i < 4; ++i) {
        const v2f a = frag_kmajor(VeT, NT, s0 + kq * 4, tbase + i * 16);
        acc[i] = wmma_f32(a, b, acc[i]);
      }
    }
    __syncthreads();
  }

  // ---- softmax over t (axis 1) for each column n = lo ----
  float m = -INFINITY;
#pragma unroll
  for (int i = 0; i < 4; ++i)
#pragma unroll
    for (int p = 0; p < 8; ++p) m = fmaxf(m, acc[i][p]);
  m = fmaxf(m, __shfl_xor(m, 16));
  if (hi == 0) red[wid][lo] = m;
  __syncthreads();
  float gm = red[0][lo];
#pragma unroll
  for (int w = 1; w < 8; ++w) gm = fmaxf(gm, red[w][lo]);
  __syncthreads();

  float s = 0.f;
#pragma unroll
  for (int i = 0; i < 4; ++i)
#pragma unroll
    for (int p = 0; p < 8; ++p) {
      acc[i][p] = __expf(acc[i][p] - gm);
      s += acc[i][p];
    }
  s += __shfl_xor(s, 16);
  if (hi == 0) red[wid][lo] = s;
  __syncthreads();
  float gs = 0.f;
#pragma unroll
  for (int w = 0; w < 8; ++w) gs += red[w][lo];
  const float inv = 1.f / gs;

  float* outc = out + (size_t)c * NT * NT + r0;
#pragma unroll
  for (int i = 0; i < 4; ++i)
#pragma unroll
    for (int p = 0; p < 8; ++p) {
      const int t = tbase + i * 16 + p + 8 * hi;
      outc[(size_t)t * NT + lo] = acc[i][p] * inv;
    }
}

// ---------------------------------------------------------------------------
extern "C" void kernel_launch(void* const* d_in, const int* in_sizes, int n_in,
                              void* d_out, int out_size, void* d_ws,
                              size_t ws_size, hipStream_t stream) {
  (void)in_sizes; (void)n_in; (void)out_size; (void)ws_size;
  const float* x   = (const float*)d_in[0];
  const float* U1  = (const float*)d_in[1];
  const float* U2  = (const float*)d_in[2];
  const float* U3  = (const float*)d_in[3];
  const float* b_e = (const float*)d_in[4];
  const float* Ve  = (const float*)d_in[5];
  float* out = (float*)d_out;

  // Workspace carve-up (~22 MB of floats)
  float* ws    = (float*)d_ws;
  float* tmpP  = ws;                                   // 64 * NC*NT
  float* tmpCT = tmpP + (size_t)64 * NC * NT;          // NC*NT
  float* Rvt   = tmpCT + (size_t)NC * NT;              // NV*NT
  float* Lvt   = Rvt + (size_t)NV * NT;                // NV*NT
  float* VeT   = Lvt + (size_t)NV * NT;                // NT*NT
  float* prod  = VeT + (size_t)NT * NT;                // NT*NT

  k1_reduce_x<<<64, 256, 0, stream>>>(x, U1, U3, tmpP, Rvt);
  k1b_sum_partials<<<(NC * NT) / 256, 256, 0, stream>>>(tmpP, tmpCT);
  k2_lhs<<<NV, 256, 0, stream>>>(U2, tmpCT, Lvt);
  k3_transpose<<<dim3(NT / 32, NT / 32), dim3(32, 8), 0, stream>>>(Ve, VeT);
  k4_product<<<32, 256, 0, stream>>>(Rvt, Lvt, prod);
  k5_attn<<<dim3(NT / 16, NC), 256, 0, stream>>>(VeT, prod, b_e, out);
}